// DeformConv1d_84739704750225
// MI455X (gfx1250) — compile-verified
//
#include <hip/hip_runtime.h>
#include <math.h>

typedef __attribute__((ext_vector_type(2))) float v2f;
typedef __attribute__((ext_vector_type(8))) float v8f;

#define NB    4
#define LSEQ  4096
#define CH    512
#define GRP   4
#define KTAP  7
#define GCH   128
#define GK    28          // GRP*KTAP
#define LN_EPS 1e-5f

// ---------------------------------------------------------------------------
// GEMM: D[M,N] = A[M,K] @ W[N,K]^T + bias[N]   (fp32, V_WMMA_F32_16X16X4_F32)
// Wave tile 64x64 (16 accum tiles), 4 waves/block arranged 2x2 -> 128x128.
// Fragment loads are single b64 per lane because both A and W are row-major
// over K: lane(r,h) needs consecutive pair at column k0+2h of row base+r.
// ---------------------------------------------------------------------------
__global__ __launch_bounds__(128) void wmma_gemm_nt(
    const float* __restrict__ A, const float* __restrict__ W,
    const float* __restrict__ bias, float* __restrict__ D,
    int M, int Kdim, int Ncol)
{
  const int lane = threadIdx.x & 31;
  const int wave = threadIdx.x >> 5;
  const int r    = lane & 15;
  const int h    = lane >> 4;
  const int m0   = blockIdx.x * 128 + (wave & 1) * 64;
  const int n0   = blockIdx.y * 128 + (wave >> 1) * 64;

  v8f acc[4][4];
#pragma unroll
  for (int j = 0; j < 4; ++j) {
    const float bv = bias[n0 + j * 16 + r];
    v8f binit;
#pragma unroll
    for (int v = 0; v < 8; ++v) binit[v] = bv;
#pragma unroll
    for (int i = 0; i < 4; ++i) acc[i][j] = binit;
  }

  const int colk = 2 * h;
  for (int k0 = 0; k0 < Kdim; k0 += 4) {
    v2f af[4], bf[4];
#pragma unroll
    for (int i = 0; i < 4; ++i)
      af[i] = *(const v2f*)(A + (size_t)(m0 + i * 16 + r) * Kdim + k0 + colk);
#pragma unroll
    for (int j = 0; j < 4; ++j)
      bf[j] = *(const v2f*)(W + (size_t)(n0 + j * 16 + r) * Kdim + k0 + colk);
#pragma unroll
    for (int i = 0; i < 4; ++i)
#pragma unroll
      for (int j = 0; j < 4; ++j)
        acc[i][j] = __builtin_amdgcn_wmma_f32_16x16x4_f32(
            false, af[i], false, bf[j], (short)0, acc[i][j], false, false);
  }

  // C/D layout: VGPR v -> row M = v + 8*h, lane r -> column N = r
#pragma unroll
  for (int i = 0; i < 4; ++i)
#pragma unroll
    for (int j = 0; j < 4; ++j)
#pragma unroll
      for (int v = 0; v < 8; ++v)
        D[(size_t)(m0 + i * 16 + 8 * h + v) * Ncol + (n0 + j * 16 + r)] =
            acc[i][j][v];
}

// ---------------------------------------------------------------------------
// Fused per-row kernel: depthwise conv(k=3,pad=1) + LayerNorm + exact GELU +
// offset/mask head projections (512 -> 28 each) + softmax + interpolation
// weight/index tables. One 256-thread block per (n,l) row.
// ---------------------------------------------------------------------------
__global__ __launch_bounds__(256) void rownet_kernel(
    const float* __restrict__ x,
    const float* __restrict__ dw_w, const float* __restrict__ dw_b,
    const float* __restrict__ ln_g, const float* __restrict__ ln_b,
    const float* __restrict__ w_off, const float* __restrict__ b_off,
    const float* __restrict__ w_mask, const float* __restrict__ b_mask,
    int* __restrict__ pf_out, int* __restrict__ pc_out,
    float* __restrict__ wf_out, float* __restrict__ wc_out)
{
  __shared__ float s_xdw[CH];
  __shared__ float s_sum[256];
  __shared__ float s_sq[256];
  __shared__ float s_head[2 * GK];

  const int row = blockIdx.x;
  const int l   = row & (LSEQ - 1);
  const int tid = threadIdx.x;
  const int c0  = tid * 2;

  // depthwise conv, k=3, zero pad
  const float* xc = x + (size_t)row * CH;
  const float2 vc = *(const float2*)(xc + c0);
  float2 vm = make_float2(0.f, 0.f), vp = make_float2(0.f, 0.f);
  if (l > 0)        vm = *(const float2*)(xc - CH + c0);
  if (l < LSEQ - 1) vp = *(const float2*)(xc + CH + c0);

  float d0 = vm.x * dw_w[c0 * 3 + 0] + vc.x * dw_w[c0 * 3 + 1] +
             vp.x * dw_w[c0 * 3 + 2] + dw_b[c0];
  float d1 = vm.y * dw_w[(c0 + 1) * 3 + 0] + vc.y * dw_w[(c0 + 1) * 3 + 1] +
             vp.y * dw_w[(c0 + 1) * 3 + 2] + dw_b[c0 + 1];

  // LayerNorm statistics (block tree reduction)
  s_sum[tid] = d0 + d1;
  s_sq[tid]  = d0 * d0 + d1 * d1;
  __syncthreads();
  for (int s = 128; s > 0; s >>= 1) {
    if (tid < s) { s_sum[tid] += s_sum[tid + s]; s_sq[tid] += s_sq[tid + s]; }
    __syncthreads();
  }
  const float mu   = s_sum[0] * (1.0f / CH);
  const float var  = s_sq[0] * (1.0f / CH) - mu * mu;
  const float rstd = rsqrtf(var + LN_EPS);

  float t0 = (d0 - mu) * rstd * ln_g[c0]     + ln_b[c0];
  float t1 = (d1 - mu) * rstd * ln_g[c0 + 1] + ln_b[c0 + 1];
  t0 = 0.5f * t0 * (1.0f + erff(t0 * 0.70710678118654752f));  // exact GELU
  t1 = 0.5f * t1 * (1.0f + erff(t1 * 0.70710678118654752f));
  s_xdw[c0] = t0;
  s_xdw[c0 + 1] = t1;
  __syncthreads();

  // 56 head dot-products (28 offset + 28 mask); 4 lanes cooperate per output
  if (tid < 224) {
    const int j = tid >> 2;
    const int q = tid & 3;
    const float* wr = (j < GK) ? (w_off + (size_t)j * CH)
                               : (w_mask + (size_t)(j - GK) * CH);
    float s = 0.f;
    for (int c = q; c < CH; c += 4) s += s_xdw[c] * wr[c];
    s += __shfl_xor(s, 1, 32);      // wave32 cross-lane reduce within quad
    s += __shfl_xor(s, 2, 32);
    if (q == 0) s_head[j] = s + ((j < GK) ? b_off[j] : b_mask[j]);
  }
  __syncthreads();

  // softmax over K per group + deformable positions/weights
  if (tid < GK) {
    const int g = tid / KTAP;
    const int k = tid - g * KTAP;

    float mmax = -1e30f;
    for (int kk = 0; kk < KTAP; ++kk)
      mmax = fmaxf(mmax, s_head[GK + g * KTAP + kk]);
    float msum = 0.f;
    for (int kk = 0; kk < KTAP; ++kk)
      msum += expf(s_head[GK + g * KTAP + kk] - mmax);
    const float mask = expf(s_head[GK + tid] - mmax) / msum;

    const float off     = s_head[tid] * 2.0f;  // OFFSET_SCALE
    const float abs_pos = (float)l + (float)(k - 3) + off;
    const float apc     = fminf(fmaxf(abs_pos, 0.f), (float)(LSEQ - 1));
    int pf = (int)apc;                  // apc >= 0 -> trunc == floor
    if (pf > LSEQ - 1) pf = LSEQ - 1;
    int pc = pf + 1;
    if (pc > LSEQ - 1) pc = LSEQ - 1;
    const float wce = apc - (float)pf;
    const float wfl = 1.f - wce;
    const float valid =
        (abs_pos >= 0.f && abs_pos <= (float)(LSEQ - 1)) ? 1.f : 0.f;

    const size_t idx = (size_t)row * GK + tid;   // [N,L,G,K]
    pf_out[idx] = pf;
    pc_out[idx] = pc;
    wf_out[idx] = wfl * valid * mask;
    wc_out[idx] = wce * valid * mask;
  }
}

// ---------------------------------------------------------------------------
// Gather: gathered[n,l,c] = sum_k wf*x_proj[n,pf,c] + wc*x_proj[n,pc,c]
// where (pf,pc,wf,wc) depend on (n,l,g=c/128,k). Channel-contiguous loads are
// fully coalesced within a wave. 256 threads cover half the channel dim.
// ---------------------------------------------------------------------------
__global__ __launch_bounds__(256) void gather_kernel(
    const float* __restrict__ x_proj,
    const int* __restrict__ pf_in, const int* __restrict__ pc_in,
    const float* __restrict__ wf_in, const float* __restrict__ wc_in,
    float* __restrict__ gathered)
{
  const int half = blockIdx.x & 1;
  const int row  = blockIdx.x >> 1;
  const int n    = row >> 12;               // row / LSEQ
  const int c    = half * 256 + threadIdx.x;
  const int g    = c >> 7;                  // c / GCH

  const float* xb  = x_proj + ((size_t)n * LSEQ) * CH;
  const size_t base = (size_t)row * GK + g * KTAP;

  float acc = 0.f;
#pragma unroll
  for (int k = 0; k < KTAP; ++k) {
    const int   pf = pf_in[base + k];
    const int   pc = pc_in[base + k];
    const float wf = wf_in[base + k];
    const float wc = wc_in[base + k];
    acc += wf * xb[(size_t)pf * CH + c] + wc * xb[(size_t)pc * CH + c];
  }
  gathered[(size_t)row * CH + c] = acc;
}

// ---------------------------------------------------------------------------
extern "C" void kernel_launch(void* const* d_in, const int* in_sizes, int n_in,
                              void* d_out, int out_size, void* d_ws,
                              size_t ws_size, hipStream_t stream)
{
  const float* x      = (const float*)d_in[0];
  const float* dw_w   = (const float*)d_in[1];
  const float* dw_b   = (const float*)d_in[2];
  const float* ln_g   = (const float*)d_in[3];
  const float* ln_b   = (const float*)d_in[4];
  const float* w_off  = (const float*)d_in[5];
  const float* b_off  = (const float*)d_in[6];
  const float* w_mask = (const float*)d_in[7];
  const float* b_mask = (const float*)d_in[8];
  const float* w_in   = (const float*)d_in[9];
  const float* b_in   = (const float*)d_in[10];
  const float* w_out  = (const float*)d_in[11];
  const float* b_out  = (const float*)d_in[12];
  float* out = (float*)d_out;

  const size_t rowsTot = (size_t)NB * LSEQ;  // 16384

  char* ws = (char*)d_ws;
  float* x_proj   = (float*)ws;  ws += rowsTot * CH * sizeof(float);   // 32 MB
  float* gathered = (float*)ws;  ws += rowsTot * CH * sizeof(float);   // 32 MB
  int*   pf = (int*)ws;          ws += rowsTot * GK * sizeof(int);
  int*   pc = (int*)ws;          ws += rowsTot * GK * sizeof(int);
  float* wf = (float*)ws;        ws += rowsTot * GK * sizeof(float);
  float* wc = (float*)ws;        ws += rowsTot * GK * sizeof(float);

  dim3 gemm_grid((unsigned)(rowsTot / 128), CH / 128);

  // 1) x_proj = x @ w_in^T + b_in
  wmma_gemm_nt<<<gemm_grid, 128, 0, stream>>>(x, w_in, b_in, x_proj,
                                              (int)rowsTot, CH, CH);
  // 2) conv + LN + GELU + heads + softmax -> interpolation tables
  rownet_kernel<<<(unsigned)rowsTot, 256, 0, stream>>>(
      x, dw_w, dw_b, ln_g, ln_b, w_off, b_off, w_mask, b_mask, pf, pc, wf, wc);
  // 3) deformable gather
  gather_kernel<<<(unsigned)(rowsTot * 2), 256, 0, stream>>>(
      x_proj, pf, pc, wf, wc, gathered);
  // 4) out = gathered @ w_out^T + b_out
  wmma_gemm_nt<<<gemm_grid, 128, 0, stream>>>(gathered, w_out, b_out, out,
                                              (int)rowsTot, CH, CH);
}